// GCN2L_89807766159535
// MI455X (gfx1250) — compile-verified
//
#include <hip/hip_runtime.h>
#include <hip/hip_bf16.h>

typedef __attribute__((ext_vector_type(2))) float v2f;
typedef __attribute__((ext_vector_type(8))) float v8f;

#define IN_CH  128
#define HID    64
#define OUTC   40
#define OUTP   48   // padded hidden-2 stride

// ---------------------------------------------------------------------------
// 1) init: deg = 1.0 (self loop), agg1[i*64+j] = b1[j], out[i*40+j] = b2[j]
// ---------------------------------------------------------------------------
__global__ void gcn_init(float* __restrict__ deg, float* __restrict__ agg1,
                         float* __restrict__ out, const float* __restrict__ b1,
                         const float* __restrict__ b2, int N) {
  int tid = blockIdx.x * blockDim.x + threadIdx.x;
  long long total = (long long)N * HID;
  if (tid < N) deg[tid] = 1.0f;
  if ((long long)tid < total) agg1[tid] = b1[tid & (HID - 1)];
  if ((long long)tid < (long long)N * OUTC) out[tid] = b2[tid % OUTC];
}

// ---------------------------------------------------------------------------
// 2) degree accumulation over real edges (self loops pre-counted in init)
// ---------------------------------------------------------------------------
__global__ void gcn_degree(const int* __restrict__ col, float* __restrict__ deg,
                           int E) {
  int e = blockIdx.x * blockDim.x + threadIdx.x;
  if (e < E) atomicAdd(&deg[col[e]], 1.0f);
}

// ---------------------------------------------------------------------------
// 3) deg -> D^-1/2 in place
// ---------------------------------------------------------------------------
__global__ void gcn_dinv(float* __restrict__ deg, int N) {
  int i = blockIdx.x * blockDim.x + threadIdx.x;
  if (i < N) {
    float d = deg[i];
    deg[i] = (d > 0.0f) ? rsqrtf(d) : 0.0f;
  }
}

// ---------------------------------------------------------------------------
// 4) GEMM1: h1[N,64] = x[N,128] @ W1[128,64]  (exact f32 WMMA 16x16x4)
//    block = 128 threads = 4 waves; block owns 16 rows; wave w owns cols 16w..
// ---------------------------------------------------------------------------
__global__ void gcn_gemm1(const float* __restrict__ x,
                          const float* __restrict__ W1,
                          float* __restrict__ h1) {
  const int wave = threadIdx.x >> 5;
  const int lane = threadIdx.x & 31;
  const int m0 = blockIdx.x * 16;
  const int n0 = wave * 16;
  const int rowA = lane & 15;          // M index within tile
  const int colB = lane & 15;          // N index within tile
  const int kOff = (lane < 16) ? 0 : 2;

  const float* xrow = x + (size_t)(m0 + rowA) * IN_CH;
  v8f acc = {};
#pragma unroll
  for (int k = 0; k < IN_CH; k += 4) {
    v2f a, b;
    a.x = xrow[k + kOff];
    a.y = xrow[k + kOff + 1];
    b.x = W1[(size_t)(k + kOff) * HID + n0 + colB];
    b.y = W1[(size_t)(k + kOff + 1) * HID + n0 + colB];
    acc = __builtin_amdgcn_wmma_f32_16x16x4_f32(false, a, false, b,
                                                (short)0, acc, false, false);
  }
  // D layout: VGPR j holds row m0 + j + (lane<16 ? 0 : 8), col n0 + (lane&15)
  const int mBase = m0 + ((lane < 16) ? 0 : 8);
  const int c = n0 + colB;
#pragma unroll
  for (int j = 0; j < 8; ++j)
    h1[(size_t)(mBase + j) * HID + c] = acc[j];
}

// ---------------------------------------------------------------------------
// 5) edge scatter layer 1: agg1[c] += h1[r] * dinv[r]*dinv[c]
//    one edge per 32 lanes, 2 channels per lane; virtual edges e>=E are
//    the self loops (r=c=e-E).
// ---------------------------------------------------------------------------
__global__ void gcn_scatter1(const int* __restrict__ row,
                             const int* __restrict__ col,
                             const float* __restrict__ dinv,
                             const float* __restrict__ h1,
                             float* __restrict__ agg1, int E, int N) {
  int idx = blockIdx.x * blockDim.x + threadIdx.x;
  int e = idx >> 5;
  int lane = idx & 31;
  if (e >= E + N) return;
  int r, c;
  if (e < E) { r = row[e]; c = col[e]; } else { r = c = e - E; }
  float nrm = dinv[r] * dinv[c];
  const float2 hv = *(const float2*)(h1 + (size_t)r * HID + lane * 2);
  float* dst = agg1 + (size_t)c * HID + lane * 2;
  atomicAdd(dst,     hv.x * nrm);
  atomicAdd(dst + 1, hv.y * nrm);
}

// ---------------------------------------------------------------------------
// 6) GEMM2: h2[N,48] = relu(agg1[N,64]) @ W2[64,40] (cols >= 40 skipped)
//    block = 96 threads = 3 waves (3 N-tiles of 16 cover 48 >= 40 cols)
// ---------------------------------------------------------------------------
__global__ void gcn_gemm2(const float* __restrict__ agg1,
                          const float* __restrict__ W2,
                          float* __restrict__ h2) {
  const int wave = threadIdx.x >> 5;   // 0..2
  const int lane = threadIdx.x & 31;
  const int m0 = blockIdx.x * 16;
  const int n0 = wave * 16;
  const int rowA = lane & 15;
  const int colB = n0 + (lane & 15);
  const int kOff = (lane < 16) ? 0 : 2;
  const bool valid = (colB < OUTC);

  const float* arow = agg1 + (size_t)(m0 + rowA) * HID;
  v8f acc = {};
#pragma unroll
  for (int k = 0; k < HID; k += 4) {
    v2f a, b;
    a.x = fmaxf(arow[k + kOff], 0.0f);       // fused ReLU
    a.y = fmaxf(arow[k + kOff + 1], 0.0f);
    b.x = valid ? W2[(size_t)(k + kOff) * OUTC + colB] : 0.0f;
    b.y = valid ? W2[(size_t)(k + kOff + 1) * OUTC + colB] : 0.0f;
    acc = __builtin_amdgcn_wmma_f32_16x16x4_f32(false, a, false, b,
                                                (short)0, acc, false, false);
  }
  const int mBase = m0 + ((lane < 16) ? 0 : 8);
  if (valid) {
#pragma unroll
    for (int j = 0; j < 8; ++j)
      h2[(size_t)(mBase + j) * OUTP + colB] = acc[j];
  }
}

// ---------------------------------------------------------------------------
// 7) edge scatter layer 2: out[c,0:40] += h2[r,0:40] * dinv[r]*dinv[c]
//    lanes 0..19 each do 2 channels
// ---------------------------------------------------------------------------
__global__ void gcn_scatter2(const int* __restrict__ row,
                             const int* __restrict__ col,
                             const float* __restrict__ dinv,
                             const float* __restrict__ h2,
                             float* __restrict__ out, int E, int N) {
  int idx = blockIdx.x * blockDim.x + threadIdx.x;
  int e = idx >> 5;
  int lane = idx & 31;
  if (e >= E + N || lane >= OUTC / 2) return;
  int r, c;
  if (e < E) { r = row[e]; c = col[e]; } else { r = c = e - E; }
  float nrm = dinv[r] * dinv[c];
  const float2 hv = *(const float2*)(h2 + (size_t)r * OUTP + lane * 2);
  float* dst = out + (size_t)c * OUTC + lane * 2;
  atomicAdd(dst,     hv.x * nrm);
  atomicAdd(dst + 1, hv.y * nrm);
}

// ---------------------------------------------------------------------------
extern "C" void kernel_launch(void* const* d_in, const int* in_sizes, int n_in,
                              void* d_out, int out_size, void* d_ws, size_t ws_size,
                              hipStream_t stream) {
  const float* x  = (const float*)d_in[0];
  const int*   ei = (const int*)d_in[1];
  const float* W1 = (const float*)d_in[2];
  const float* b1 = (const float*)d_in[3];
  const float* W2 = (const float*)d_in[4];
  const float* b2 = (const float*)d_in[5];
  float* out = (float*)d_out;

  const int N = in_sizes[0] / IN_CH;   // 100000
  const int E = in_sizes[1] / 2;       // 1600000
  const int* rowp = ei;                // ei[0] = sources
  const int* colp = ei + E;            // ei[1] = targets

  float* ws   = (float*)d_ws;
  float* deg  = ws;                                   // N floats (becomes dinv)
  float* h1   = ws + N;                               // N*64 floats (reused as h2)
  float* agg1 = ws + N + (size_t)N * HID;             // N*64 floats

  // 1) init
  {
    long long total = (long long)N * HID;
    int blocks = (int)((total + 255) / 256);
    gcn_init<<<blocks, 256, 0, stream>>>(deg, agg1, out, b1, b2, N);
  }
  // 2) degrees
  gcn_degree<<<(E + 255) / 256, 256, 0, stream>>>(colp, deg, E);
  // 3) D^-1/2
  gcn_dinv<<<(N + 255) / 256, 256, 0, stream>>>(deg, N);
  // 4) GEMM1 (f32 WMMA)
  gcn_gemm1<<<N / 16, 128, 0, stream>>>(x, W1, h1);
  // 5) scatter layer 1 (edges + self loops)
  {
    long long work = (long long)(E + N) * 32;
    int blocks = (int)((work + 255) / 256);
    gcn_scatter1<<<blocks, 256, 0, stream>>>(rowp, colp, deg, h1, agg1, E, N);
  }
  // 6) GEMM2 (f32 WMMA, fused ReLU) — h2 reuses h1's workspace
  gcn_gemm2<<<N / 16, 96, 0, stream>>>(agg1, W2, h1);
  // 7) scatter layer 2 into d_out
  {
    long long work = (long long)(E + N) * 32;
    int blocks = (int)((work + 255) / 256);
    gcn_scatter2<<<blocks, 256, 0, stream>>>(rowp, colp, deg, h1, out, E, N);
  }
}